// involution_71012989272344
// MI455X (gfx1250) — compile-verified
//
#include <hip/hip_runtime.h>
#include <hip/hip_bf16.h>

// ---------------------------------------------------------------------------
// Problem constants (fixed by the reference)
// ---------------------------------------------------------------------------
#define Bsz   4
#define Cin   256
#define CRdim 64          // Cin / 4
#define Gdim  16          // groups = Cin / 16
#define GCdim 16          // group channels
#define Kker  7
#define KK    49          // Kker*Kker
#define M2    784         // KK * Gdim  (= 49*16, divisible by 16 but not 32)
#define OUTC  256
#define Hdim  64
#define Wdim  64
#define HW    4096        // Hdim*Wdim
#define NWAVES 4          // waves per block for GEMM kernels

typedef __attribute__((ext_vector_type(2))) float v2f;
typedef __attribute__((ext_vector_type(8))) float v8f;

// ---------------------------------------------------------------------------
// Wave-level (16*NTM)M x (16*NTN)N fp32 GEMM tile via V_WMMA_F32_16X16X4_F32.
// A row-major MxK (lda), B row-major KxN (ldb). A/B fragments loaded once per
// K-step and reused across the NTM x NTN wmma issues (register blocking).
// ISA 16x4 f32 A layout: lanes 0-15: M=lane, v0=K0/v1=K1;
//                        lanes 16-31: M=lane-16, v0=K2/v1=K3. B mirrors in N.
// C/D: v[r] = row (r + 8*half), col = lane%16.
// ---------------------------------------------------------------------------
template <int NTM, int NTN>
__device__ __forceinline__ void wmma_gemm_block(const float* __restrict__ A, int lda,
                                                const float* __restrict__ Bm, int ldb,
                                                int Ksz, int mBase, int nBase,
                                                v8f acc[NTM][NTN]) {
  const int lane = threadIdx.x & 31;
  const int half = lane >> 4;
  const int l16  = lane & 15;
  const float* arow = A + (size_t)(mBase + l16) * lda + 2 * half;
  const float* bptr = Bm + (size_t)(2 * half) * ldb + nBase + l16;
  for (int k = 0; k < Ksz; k += 4) {
    v2f a[NTM];
#pragma unroll
    for (int tm = 0; tm < NTM; ++tm) {
      a[tm].x = arow[(size_t)tm * 16 * lda + k + 0];
      a[tm].y = arow[(size_t)tm * 16 * lda + k + 1];
    }
#pragma unroll
    for (int tn = 0; tn < NTN; ++tn) {
      v2f b;
      b.x = bptr[(size_t)(k + 0) * ldb + tn * 16];
      b.y = bptr[(size_t)(k + 1) * ldb + tn * 16];
#pragma unroll
      for (int tm = 0; tm < NTM; ++tm) {
        // 8 args: (neg_a, A, neg_b, B, c_mod, C, reuse_a, reuse_b)
        acc[tm][tn] = __builtin_amdgcn_wmma_f32_16x16x4_f32(
            false, a[tm], false, b, (short)0, acc[tm][tn], false, false);
      }
    }
  }
}

// ---------------------------------------------------------------------------
// Kernel 1: y = relu(BN(w1 @ x))   per batch;  M=64, K=256, N=4096
// Wave tile 32M x 32N (NTM=2, NTN=2); block covers 32M x 128N.
// ---------------------------------------------------------------------------
__global__ void k_gemm_bn_relu(const float* __restrict__ x,
                               const float* __restrict__ w1,
                               const float* __restrict__ gamma,
                               const float* __restrict__ beta,
                               const float* __restrict__ mean,
                               const float* __restrict__ var,
                               float* __restrict__ y) {
  const int wave  = threadIdx.x >> 5;
  const int nBase = (blockIdx.x * NWAVES + wave) * 32;
  const int mBase = blockIdx.y * 32;
  const int b     = blockIdx.z;
  const float* Bm = x + (size_t)b * Cin * HW;
  v8f acc[2][2] = {};
  wmma_gemm_block<2, 2>(w1, Cin, Bm, HW, Cin, mBase, nBase, acc);

  float* Yb = y + (size_t)b * CRdim * HW;
  const int lane = threadIdx.x & 31;
  const int half = lane >> 4, l16 = lane & 15;
#pragma unroll
  for (int tm = 0; tm < 2; ++tm) {
#pragma unroll
    for (int r = 0; r < 8; ++r) {
      const int m = mBase + tm * 16 + r + 8 * half;
      const float s  = gamma[m] * rsqrtf(var[m] + 1e-5f);
      const float mu = mean[m], bt = beta[m];
#pragma unroll
      for (int tn = 0; tn < 2; ++tn) {
        const float v = (acc[tm][tn][r] - mu) * s + bt;
        Yb[(size_t)m * HW + nBase + tn * 16 + l16] = fmaxf(v, 0.0f);
      }
    }
  }
}

// ---------------------------------------------------------------------------
// Kernel 2: wt = w2 @ y + b2   per batch;  M=784, K=64, N=4096
// M = 49*16 not divisible by 32 -> 16M x 64N strip (NTM=1, NTN=4).
// ---------------------------------------------------------------------------
__global__ void k_gemm_bias(const float* __restrict__ y,
                            const float* __restrict__ w2,
                            const float* __restrict__ b2,
                            float* __restrict__ wt) {
  const int wave  = threadIdx.x >> 5;
  const int nBase = (blockIdx.x * NWAVES + wave) * 64;
  const int mBase = blockIdx.y * 16;
  const int b     = blockIdx.z;
  const float* Bm = y + (size_t)b * CRdim * HW;
  v8f acc[1][4] = {};
  wmma_gemm_block<1, 4>(w2, CRdim, Bm, HW, CRdim, mBase, nBase, acc);

  float* Wb = wt + (size_t)b * M2 * HW;
  const int lane = threadIdx.x & 31;
  const int half = lane >> 4, l16 = lane & 15;
#pragma unroll
  for (int r = 0; r < 8; ++r) {
    const int m = mBase + r + 8 * half;
    const float bias = b2[m];
#pragma unroll
    for (int tn = 0; tn < 4; ++tn) {
      Wb[(size_t)m * HW + nBase + tn * 16 + l16] = acc[0][tn][r] + bias;
    }
  }
}

// ---------------------------------------------------------------------------
// Kernel 3: involution stencil with LDS tap sharing.
//   inv[b,c,y,x] = sum_{k} wt[b, (c/16)*49+k, y, x] * x_pad[b,c,y+dy-3,x+dx-3]
// Block = 256 threads = 16 channels x 16 x-pixels at fixed (b, g, y).
// The 49x16 wt tap tile is staged in LDS once (coalesced global loads) and
// each tap is then a conflict-free LDS broadcast across the 16 channel lanes.
// ---------------------------------------------------------------------------
__global__ void k_involution(const float* __restrict__ x,
                             const float* __restrict__ wt,
                             float* __restrict__ inv) {
  __shared__ float taps[KK * 16];          // 3136 B

  const int xtile = blockIdx.x & 3;        // W/16 = 4 tiles
  const int py    = blockIdx.x >> 2;       // row 0..63
  const int g     = blockIdx.y;
  const int b     = blockIdx.z;
  const int tid   = threadIdx.x;
  const int pxi   = tid & 15;
  const int ch    = tid >> 4;
  const int xbase = xtile * 16;

  // Stage wt taps: element e = k*16 + pxi  ->  wt[b, g*49+k, py, xbase+pxi]
  const float* wtp = wt + ((size_t)b * M2 + (size_t)g * KK) * HW + py * Wdim + xbase;
  for (int e = tid; e < KK * 16; e += 256) {
    const int k = e >> 4;
    taps[e] = wtp[(size_t)k * HW + (e & 15)];
  }
  __syncthreads();

  const int c = g * GCdim + ch;
  const float* xp = x + ((size_t)b * Cin + c) * HW;
  const int px = xbase + pxi;

  float acc = 0.0f;
#pragma unroll
  for (int dy = 0; dy < Kker; ++dy) {
    const int yy = py + dy - 3;
    const bool okY = (unsigned)yy < (unsigned)Hdim;
#pragma unroll
    for (int dx = 0; dx < Kker; ++dx) {
      const int xx = px + dx - 3;
      const bool ok = okY && ((unsigned)xx < (unsigned)Wdim);
      const float xv = ok ? xp[yy * Wdim + xx] : 0.0f;
      const float wv = taps[(dy * Kker + dx) * 16 + pxi];
      acc = fmaf(wv, xv, acc);
    }
  }
  inv[((size_t)b * Cin + c) * HW + py * Wdim + px] = acc;
}

// ---------------------------------------------------------------------------
// Kernel 4: out = wp @ inv   per batch;  M=256, K=256, N=4096
// Wave tile 32M x 32N (NTM=2, NTN=2); halves B-panel L2 traffic vs 16M strip.
// ---------------------------------------------------------------------------
__global__ void k_gemm_plain(const float* __restrict__ inv,
                             const float* __restrict__ wp,
                             float* __restrict__ out) {
  const int wave  = threadIdx.x >> 5;
  const int nBase = (blockIdx.x * NWAVES + wave) * 32;
  const int mBase = blockIdx.y * 32;
  const int b     = blockIdx.z;
  const float* Bm = inv + (size_t)b * Cin * HW;
  v8f acc[2][2] = {};
  wmma_gemm_block<2, 2>(wp, Cin, Bm, HW, Cin, mBase, nBase, acc);

  float* Ob = out + (size_t)b * OUTC * HW;
  const int lane = threadIdx.x & 31;
  const int half = lane >> 4, l16 = lane & 15;
#pragma unroll
  for (int tm = 0; tm < 2; ++tm) {
#pragma unroll
    for (int r = 0; r < 8; ++r) {
      const int m = mBase + tm * 16 + r + 8 * half;
#pragma unroll
      for (int tn = 0; tn < 2; ++tn) {
        Ob[(size_t)m * HW + nBase + tn * 16 + l16] = acc[tm][tn][r];
      }
    }
  }
}

// ---------------------------------------------------------------------------
// Launch
// ---------------------------------------------------------------------------
extern "C" void kernel_launch(void* const* d_in, const int* in_sizes, int n_in,
                              void* d_out, int out_size, void* d_ws, size_t ws_size,
                              hipStream_t stream) {
  const float* x     = (const float*)d_in[0];
  const float* w1    = (const float*)d_in[1];
  const float* gamma = (const float*)d_in[2];
  const float* beta  = (const float*)d_in[3];
  const float* mean  = (const float*)d_in[4];
  const float* var   = (const float*)d_in[5];
  const float* w2    = (const float*)d_in[6];
  const float* b2    = (const float*)d_in[7];
  const float* wp    = (const float*)d_in[8];
  float* out = (float*)d_out;

  // Workspace layout (all fp32):
  //   y   : [B][64][4096]   =  4 MB
  //   wt  : [B][784][4096]  = 51 MB
  //   inv : [B][256][4096]  = 17 MB   (total ~72 MB, fits 192 MB global L2)
  float* y   = (float*)d_ws;
  float* wt  = y  + (size_t)Bsz * CRdim * HW;
  float* inv = wt + (size_t)Bsz * M2 * HW;

  dim3 blk(32 * NWAVES, 1, 1);

  dim3 g1(HW / (32 * NWAVES), CRdim / 32, Bsz);   // (32, 2, 4)
  k_gemm_bn_relu<<<g1, blk, 0, stream>>>(x, w1, gamma, beta, mean, var, y);

  dim3 g2(HW / (64 * NWAVES), M2 / 16, Bsz);      // (16, 49, 4)
  k_gemm_bias<<<g2, blk, 0, stream>>>(y, w2, b2, wt);

  dim3 gI(Hdim * (Wdim / 16), Gdim, Bsz);         // (256, 16, 4)
  k_involution<<<gI, 256, 0, stream>>>(x, wt, inv);

  dim3 g4(HW / (32 * NWAVES), OUTC / 32, Bsz);    // (32, 8, 4)
  k_gemm_plain<<<g4, blk, 0, stream>>>(inv, wp, out);
}